// TopKPool_14104672600399
// MI455X (gfx1250) — compile-verified
//
#include <hip/hip_runtime.h>
#include <hip/hip_bf16.h>
#include <math.h>

typedef __attribute__((ext_vector_type(2))) float v2f;
typedef __attribute__((ext_vector_type(8))) float v8f;

#define B_ 16
#define N_ 4096
#define F_ 128
#define K_ 1024

// ---------------------------------------------------------------------------
// Kernel 1: scores[b,n] = dot(node_embed[b,n,:], w)  via V_WMMA_F32_16X16X4_F32
// One wave per 16 nodes. A[i][k] = w[k] (broadcast rows)  => D[i][j] = score_j
// A layout (f32 16x4): lanes 0-15 hold K={0,1} in v0/v1; lanes 16-31 hold K={2,3}
// B layout (f32 4x16): v0 holds K=0 (lanes 0-15, N=lane) / K=2 (lanes 16-31),
//                      v1 holds K=1 / K=3, N = lane & 15.
// D row 0 sits in acc[0] of lanes 0-15 (N = lane).
// ---------------------------------------------------------------------------
__global__ void topk_scores_wmma(const float* __restrict__ embed,
                                 const float* __restrict__ w,
                                 float* __restrict__ scores) {
    int wave = (int)((blockIdx.x * blockDim.x + threadIdx.x) >> 5);
    int lane = (int)(threadIdx.x & 31);
    int total = (B_ * N_) / 16;
    if (wave >= total) return;                 // uniform per wave: EXEC stays all-1s
    int b  = wave / (N_ / 16);
    int n0 = (wave % (N_ / 16)) * 16;

    const float* eb = embed + ((long)b * N_ + n0) * F_;
    int half = lane >> 4;                      // 0: K base 0,1 ; 1: K base 2,3
    int nl   = lane & 15;                      // column (node) index for B operand

    v8f acc = {};
    for (int k0 = 0; k0 < F_; k0 += 4) {
        v2f a, bm;
        a[0] = w[k0 + 2 * half];
        a[1] = w[k0 + 2 * half + 1];
        bm[0] = eb[(long)nl * F_ + k0 + 2 * half];
        bm[1] = eb[(long)nl * F_ + k0 + 2 * half + 1];
        acc = __builtin_amdgcn_wmma_f32_16x16x4_f32(
            /*neg_a=*/false, a, /*neg_b=*/false, bm,
            /*c_mod=*/(short)0, acc, /*reuse_a=*/false, /*reuse_b=*/false);
    }
    if (lane < 16) scores[(long)b * N_ + n0 + lane] = acc[0];
}

// ---------------------------------------------------------------------------
// Kernel 2: per-batch bitonic sort of 4096 (score, idx) pairs in LDS.
// Order: descending score, ties -> ascending index (matches jax.lax.top_k).
// Emits top_idx[b, 0:K] and gate[b, 0:K] = tanh(top_val).
// ---------------------------------------------------------------------------
__global__ __launch_bounds__(1024)
void topk_sort(const float* __restrict__ scores,
               int* __restrict__ top_idx,
               float* __restrict__ gate) {
    __shared__ float sv[N_];
    __shared__ int   si[N_];
    int b   = (int)blockIdx.x;
    int tid = (int)threadIdx.x;

    for (int e = tid; e < N_; e += 1024) { sv[e] = scores[(long)b * N_ + e]; si[e] = e; }
    __syncthreads();

    for (int k = 2; k <= N_; k <<= 1) {
        for (int j = k >> 1; j > 0; j >>= 1) {
            for (int e = tid; e < N_; e += 1024) {
                int p = e ^ j;
                if (p > e) {
                    float va = sv[e], vb = sv[p];
                    int   ia = si[e], ib = si[p];
                    bool a_first = (va > vb) || (va == vb && ia < ib);
                    bool dir_desc = ((e & k) == 0);
                    if (dir_desc ? !a_first : a_first) {
                        sv[e] = vb; sv[p] = va;
                        si[e] = ib; si[p] = ia;
                    }
                }
            }
            __syncthreads();
        }
    }
    for (int e = tid; e < K_; e += 1024) {
        top_idx[(long)b * K_ + e] = si[e];
        gate[(long)b * K_ + e]    = tanhf(sv[e]);
    }
}

// ---------------------------------------------------------------------------
// Kernel 3: pooled_nodes[b,r,:] = embed[b, idx[b,r], :] * gate[b,r]
// One wave per row: 32 lanes x float4 = 128 floats (b128 load/store).
// ---------------------------------------------------------------------------
__global__ void gather_nodes(const float* __restrict__ embed,
                             const int* __restrict__ top_idx,
                             const float* __restrict__ gate,
                             float* __restrict__ out) {
    int wave = (int)((blockIdx.x * blockDim.x + threadIdx.x) >> 5);
    int lane = (int)(threadIdx.x & 31);
    if (wave >= B_ * K_) return;
    int b = wave / K_, r = wave % K_;
    int idx = top_idx[(long)b * K_ + r];
    float g = gate[(long)b * K_ + r];
    const float4* src = (const float4*)(embed + ((long)b * N_ + idx) * F_);
    float4*       dst = (float4*)(out + ((long)b * K_ + r) * F_);
    float4 v = src[lane];
    v.x *= g; v.y *= g; v.z *= g; v.w *= g;
    dst[lane] = v;
}

// ---------------------------------------------------------------------------
// Kernel 4: pooled_adj[b,r,c] = adj[b, idx[b,r], idx[b,c]]
// One block per output row; column indices staged once in LDS.
// ---------------------------------------------------------------------------
__global__ void gather_adj(const float* __restrict__ adj,
                           const int* __restrict__ top_idx,
                           float* __restrict__ out) {
    __shared__ int scol[K_];
    int b = (int)blockIdx.x / K_;
    int r = (int)blockIdx.x % K_;
    const int* idxb = top_idx + (long)b * K_;
    for (int c = (int)threadIdx.x; c < K_; c += (int)blockDim.x) scol[c] = idxb[c];
    __syncthreads();

    int row = scol[r];
    const float* arow = adj + ((long)b * N_ + row) * N_;
    float*       orow = out + (long)blockIdx.x * K_;
    for (int c = (int)threadIdx.x; c < K_; c += (int)blockDim.x)
        orow[c] = arow[scol[c]];
}

extern "C" void kernel_launch(void* const* d_in, const int* in_sizes, int n_in,
                              void* d_out, int out_size, void* d_ws, size_t ws_size,
                              hipStream_t stream) {
    const float* embed = (const float*)d_in[0];   // [B, N, F]
    const float* adj   = (const float*)d_in[1];   // [B, N, N]
    const float* w     = (const float*)d_in[2];   // [F]

    float* scores  = (float*)d_ws;                          // B*N floats
    int*   top_idx = (int*)(scores + (size_t)B_ * N_);      // B*K ints
    float* gate    = (float*)(top_idx + (size_t)B_ * K_);   // B*K floats

    float* out_nodes = (float*)d_out;                       // B*K*F
    float* out_adj   = out_nodes + (size_t)B_ * K_ * F_;    // B*K*K

    // 1) scores via WMMA: one wave / 16 nodes -> 4096 waves, 8 waves per block
    topk_scores_wmma<<<(B_ * N_ / 16) / 8, 256, 0, stream>>>(embed, w, scores);
    // 2) per-batch bitonic top-k
    topk_sort<<<B_, 1024, 0, stream>>>(scores, top_idx, gate);
    // 3) pooled_nodes: one wave per row, 8 rows per block
    gather_nodes<<<(B_ * K_) / 8, 256, 0, stream>>>(embed, top_idx, gate, out_nodes);
    // 4) pooled_adj: one block per row
    gather_adj<<<B_ * K_, 256, 0, stream>>>(adj, top_idx, out_adj);
}